// SCAN_26164940767405
// MI455X (gfx1250) — compile-verified
//
#include <hip/hip_runtime.h>
#include <math.h>

// Problem constants (match reference)
#define BB 64      // batch (images == captions)
#define RR 36      // regions per image
#define RP 48      // regions padded to 3x16 WMMA tiles
#define LL 64      // tokens per caption
#define EE 512     // embedding dim
#define KC 64      // K-chunk staged through LDS
#define LAMBDA 9.0f

typedef float v2f __attribute__((ext_vector_type(2)));
typedef float v8f __attribute__((ext_vector_type(8)));

// D = A(16x4 f32) * B(4x16 f32) + C(16x16 f32), exact fp32 matrix op on CDNA5.
__device__ __forceinline__ v8f wmma_f32(v2f a, v2f b, v8f c) {
  return __builtin_amdgcn_wmma_f32_16x16x4_f32(
      /*neg_a=*/false, a, /*neg_b=*/false, b,
      /*c_mod=*/(short)0, c, /*reuse_a=*/false, /*reuse_b=*/false);
}

// ---------------------------------------------------------------------------
// Kernel 1: L2-normalize every caption row (B*L) and image-region row (B*R).
// Also store na[c,l] = ||capn[c,l]|| (== norm/max(norm,eps), i.e. ~1).
// One 128-thread block per row; 4 floats per thread.
// ---------------------------------------------------------------------------
__global__ __launch_bounds__(128) void normalize_rows(
    const float* __restrict__ captions, const float* __restrict__ images,
    float* __restrict__ capn, float* __restrict__ imgn, float* __restrict__ na) {
  __shared__ float red[128];
  const int row = blockIdx.x;
  const int tid = threadIdx.x;
  const bool is_cap = row < BB * LL;
  const float* src = is_cap ? captions + (size_t)row * EE
                            : images + (size_t)(row - BB * LL) * EE;
  const float4 v = *(const float4*)(src + tid * 4);
  red[tid] = v.x * v.x + v.y * v.y + v.z * v.z + v.w * v.w;
  __syncthreads();
  for (int off = 64; off > 0; off >>= 1) {
    if (tid < off) red[tid] += red[tid + off];
    __syncthreads();
  }
  const float norm = sqrtf(red[0]);
  const float scale = 1.0f / fmaxf(norm, 1e-12f);
  float* dst = is_cap ? capn + (size_t)row * EE
                      : imgn + (size_t)(row - BB * LL) * EE;
  float4 o;
  o.x = v.x * scale; o.y = v.y * scale; o.z = v.z * scale; o.w = v.w * scale;
  *(float4*)(dst + tid * 4) = o;
  if (is_cap && tid == 0) na[row] = norm * scale;
}

// ---------------------------------------------------------------------------
// Kernel 2: per-image Gram matrix G = imgn_i * imgn_i^T, zero-padded to 48x48.
// grid(9 tiles, 64 images), 1 wave per block. K loop over E=512 in steps of 4.
// B fragment uses G symmetry: B[k][n] = imgn[n][k] -> row-gather like A.
// ---------------------------------------------------------------------------
__global__ __launch_bounds__(32) void gram_kernel(
    const float* __restrict__ imgn, float* __restrict__ G) {
  const int i = blockIdx.y;
  const int t = blockIdx.x, ti = t / 3, tj = t % 3;
  const int lane = threadIdx.x;
  const int hi = lane >> 4, n16 = lane & 15;
  const int ar = ti * 16 + n16, br = tj * 16 + n16;
  const bool am = ar < RR, bm = br < RR;
  const float* Arow = imgn + ((size_t)i * RR + (am ? ar : 0)) * EE;
  const float* Brow = imgn + ((size_t)i * RR + (bm ? br : 0)) * EE;
  v8f acc = {};
#pragma unroll 8
  for (int k = 0; k < EE; k += 4) {
    const int kk = k + 2 * hi;
    v2f a = *(const v2f*)(Arow + kk);
    v2f b = *(const v2f*)(Brow + kk);
    if (!am) { a.x = 0.f; a.y = 0.f; }
    if (!bm) { b.x = 0.f; b.y = 0.f; }
    acc = wmma_f32(a, b, acc);
  }
#pragma unroll
  for (int j = 0; j < 8; ++j) {
    const int row = ti * 16 + j + 8 * hi;
    const int col = tj * 16 + n16;
    G[((size_t)i * RP + row) * RP + col] = acc[j];  // padded entries are exact zeros
  }
}

// ---------------------------------------------------------------------------
// Kernel 3: one block per (image i, caption c) pair. 128 threads = 4 waves.
// Phase A: S = capn_c (64xE) @ imgn_i^T (E x 48pad) via WMMA, K tiled via LDS.
// Phase B: row softmax(9*S) over 36 regions, dot[l] = sum_r w*s.
// Phase C: V = W @ G (48x48, symmetric) via WMMA; nb2[l] = sum_r W[l,r]*V[l,r].
// Phase D: r[l] = dot / max(na*nb, 1e-8), masked mean over L.
// ---------------------------------------------------------------------------
__global__ __launch_bounds__(128) void scan_main(
    const float* __restrict__ capn, const float* __restrict__ imgn,
    const float* __restrict__ na_g, const float* __restrict__ G,
    const int* __restrict__ cap_lens, float* __restrict__ out) {
  __shared__ float Ash[LL][KC + 2];  // 64x66 cap K-chunk (reused for W.*V later)
  __shared__ float Bsh[RP][KC + 2];  // 48x66 img K-chunk (rows>=36 zeroed)
  __shared__ float Wsh[LL][RP + 2];  // 64x50 sim then softmax weights
  __shared__ float Gsh[RP][RP + 2];  // 48x50 Gram tile
  __shared__ float dotsh[LL];
  __shared__ float rsh[LL];

  const int i = blockIdx.x, c = blockIdx.y;
  const int tid = threadIdx.x;
  const int wave = tid >> 5, lane = tid & 31;
  const int hi = lane >> 4, n16 = lane & 15;

  // stage Gram matrix (read once, reused by all 4 waves in phase C)
  for (int idx = tid; idx < RP * RP; idx += 128) {
    const int r = idx / RP, col = idx % RP;
    Gsh[r][col] = G[((size_t)i * RP + r) * RP + col];
  }

  const float* Abase = capn + (size_t)c * LL * EE;
  const float* Bbase = imgn + (size_t)i * RR * EE;

  v8f acc0 = {}, acc1 = {}, acc2 = {};
  for (int kc = 0; kc < EE; kc += KC) {
    __syncthreads();
    // stage A: 64 rows x 64 cols (1024 float4 / 128 threads)
#pragma unroll
    for (int q = 0; q < 8; ++q) {
      const int idx = tid + q * 128;
      const int row = idx >> 4, c4 = (idx & 15) << 2;
      const float4 v = *(const float4*)(Abase + row * EE + kc + c4);
      Ash[row][c4] = v.x; Ash[row][c4 + 1] = v.y;
      Ash[row][c4 + 2] = v.z; Ash[row][c4 + 3] = v.w;
      if (kc + KC < EE)
        __builtin_prefetch(Abase + row * EE + kc + KC + c4, 0, 1);
    }
    // stage B: 48 rows x 64 cols (rows >= 36 zero)
#pragma unroll
    for (int q = 0; q < 6; ++q) {
      const int idx = tid + q * 128;
      const int row = idx >> 4, c4 = (idx & 15) << 2;
      float4 v; v.x = 0.f; v.y = 0.f; v.z = 0.f; v.w = 0.f;
      if (row < RR) v = *(const float4*)(Bbase + row * EE + kc + c4);
      Bsh[row][c4] = v.x; Bsh[row][c4 + 1] = v.y;
      Bsh[row][c4 + 2] = v.z; Bsh[row][c4 + 3] = v.w;
    }
    __syncthreads();
#pragma unroll
    for (int k0 = 0; k0 < KC; k0 += 4) {
      const int kk = k0 + 2 * hi;
      const v2f a  = *(const v2f*)&Ash[wave * 16 + n16][kk];
      const v2f b0 = *(const v2f*)&Bsh[n16][kk];
      const v2f b1 = *(const v2f*)&Bsh[16 + n16][kk];
      const v2f b2 = *(const v2f*)&Bsh[32 + n16][kk];
      acc0 = wmma_f32(a, b0, acc0);
      acc1 = wmma_f32(a, b1, acc1);
      acc2 = wmma_f32(a, b2, acc2);
    }
  }

  // write S fragments into Wsh (C layout: VGPR j -> M = j + 8*hi, N = lane&15)
  __syncthreads();
#pragma unroll
  for (int j = 0; j < 8; ++j) {
    const int l = wave * 16 + j + 8 * hi;
    Wsh[l][n16]      = acc0[j];
    Wsh[l][16 + n16] = acc1[j];
    Wsh[l][32 + n16] = acc2[j];
  }
  __syncthreads();

  // softmax over 36 regions, temperature 9; dot[l] = sum_r w*s; pad cols -> 0
  if (tid < LL) {
    float m = -1e30f;
    for (int r = 0; r < RR; ++r) m = fmaxf(m, Wsh[tid][r]);
    float sum = 0.f;
    for (int r = 0; r < RR; ++r) sum += expf(LAMBDA * (Wsh[tid][r] - m));
    const float inv = 1.0f / sum;
    float dot = 0.f;
    for (int r = 0; r < RR; ++r) {
      const float s = Wsh[tid][r];
      const float w = expf(LAMBDA * (s - m)) * inv;
      dot += w * s;
      Wsh[tid][r] = w;
    }
    for (int r = RR; r < RP; ++r) Wsh[tid][r] = 0.f;
    dotsh[tid] = dot;
  }
  __syncthreads();

  // V = W (64x48) @ G (48x48, symmetric -> B fragments are row gathers)
  v8f v0 = {}, v1 = {}, v2 = {};
#pragma unroll
  for (int k0 = 0; k0 < RP; k0 += 4) {
    const int kk = k0 + 2 * hi;
    const v2f a  = *(const v2f*)&Wsh[wave * 16 + n16][kk];
    const v2f b0 = *(const v2f*)&Gsh[n16][kk];
    const v2f b1 = *(const v2f*)&Gsh[16 + n16][kk];
    const v2f b2 = *(const v2f*)&Gsh[32 + n16][kk];
    v0 = wmma_f32(a, b0, v0);
    v1 = wmma_f32(a, b1, v1);
    v2 = wmma_f32(a, b2, v2);
  }

  // elementwise W .* V into Ash (reuse), then nb2[l] row sums
  __syncthreads();
#pragma unroll
  for (int j = 0; j < 8; ++j) {
    const int l = wave * 16 + j + 8 * hi;
    Ash[l][n16]      = v0[j] * Wsh[l][n16];
    Ash[l][16 + n16] = v1[j] * Wsh[l][16 + n16];
    Ash[l][32 + n16] = v2[j] * Wsh[l][32 + n16];
  }
  __syncthreads();

  if (tid < LL) {
    float nb2 = 0.f;
    for (int r = 0; r < RR; ++r) nb2 += Ash[tid][r];
    const float nb = sqrtf(fmaxf(nb2, 0.f));
    const float na = na_g[c * LL + tid];
    const float denom = fmaxf(na * nb, 1e-8f);
    float rl = dotsh[tid] / denom;
    if (tid >= cap_lens[c]) rl = 0.f;
    rsh[tid] = rl;
  }
  __syncthreads();
  if (tid == 0) {
    float s = 0.f;
    for (int l = 0; l < LL; ++l) s += rsh[l];
    out[(size_t)c * BB + i] = s * (1.0f / LL);
  }
}

// ---------------------------------------------------------------------------
extern "C" void kernel_launch(void* const* d_in, const int* in_sizes, int n_in,
                              void* d_out, int out_size, void* d_ws, size_t ws_size,
                              hipStream_t stream) {
  (void)in_sizes; (void)n_in; (void)out_size; (void)ws_size;
  const float* images = (const float*)d_in[0];    // (B, R, E) f32
  const float* captions = (const float*)d_in[1];  // (B, L, E) f32
  const int* cap_lens = (const int*)d_in[2];      // (B,) i32
  float* out = (float*)d_out;                     // (B_cap, B_img) f32

  float* ws = (float*)d_ws;
  float* capn = ws;                                   // B*L*E
  float* imgn = capn + (size_t)BB * LL * EE;          // B*R*E
  float* na   = imgn + (size_t)BB * RR * EE;          // B*L
  float* G    = na + (size_t)BB * LL;                 // B*RP*RP

  normalize_rows<<<BB * LL + BB * RR, 128, 0, stream>>>(captions, images, capn,
                                                        imgn, na);
  gram_kernel<<<dim3(9, BB), 32, 0, stream>>>(imgn, G);
  scan_main<<<dim3(BB, BB), 128, 0, stream>>>(capn, imgn, na, G, cap_lens, out);
}